// Wnet_49830210568741
// MI455X (gfx1250) — compile-verified
//
#include <hip/hip_runtime.h>
#include <hip/hip_bf16.h>

typedef __attribute__((ext_vector_type(2))) float v2f;
typedef __attribute__((ext_vector_type(8))) float v8f;

#define WMMA_F32_K4(a, b, c) \
    __builtin_amdgcn_wmma_f32_16x16x4_f32(false, (a), false, (b), (short)0, (c), false, false)

__device__ __forceinline__ int reflect_idx(int p, int n) {
    if (p < 0) p = -p;
    else if (p >= n) p = 2 * n - 2 - p;
    return p;
}

// ---------------------------------------------------------------------------
// Encoder block: reflect-pad(2), 5x5 conv stride 2, eval BatchNorm, ReLU.
// Direct fp32 conv; weights + folded BN affine staged in LDS.
// ---------------------------------------------------------------------------
template <int CIN, int COUT>
__global__ void __launch_bounds__(256) enc_block_kernel(
    const float* __restrict__ in, const float* __restrict__ w,
    const float* __restrict__ bias, const float* __restrict__ bng,
    const float* __restrict__ bnb, const float* __restrict__ bnm,
    const float* __restrict__ bnv, float* __restrict__ out,
    int B, int Hin, int Hout) {
    __shared__ float ws[COUT * CIN * 25];
    __shared__ float As[COUT];
    __shared__ float Bs[COUT];
    for (int i = threadIdx.x; i < COUT * CIN * 25; i += blockDim.x) ws[i] = w[i];
    if (threadIdx.x < COUT) {
        int c = threadIdx.x;
        float sc = bng[c] * rsqrtf(bnv[c] + 1e-5f);
        As[c] = sc;
        Bs[c] = (bias[c] - bnm[c]) * sc + bnb[c];
    }
    __syncthreads();

    int gid = blockIdx.x * blockDim.x + threadIdx.x;
    int total = B * Hout * Hout;
    if (gid >= total) return;
    int ox = gid % Hout;
    int oy = (gid / Hout) % Hout;
    int b  = gid / (Hout * Hout);

    float acc[COUT];
#pragma unroll
    for (int o = 0; o < COUT; ++o) acc[o] = 0.f;

    int ixs[5], iys[5];
#pragma unroll
    for (int k = 0; k < 5; ++k) {
        iys[k] = reflect_idx(2 * oy - 2 + k, Hin);
        ixs[k] = reflect_idx(2 * ox - 2 + k, Hin);
    }

    for (int c = 0; c < CIN; ++c) {
        const float* inp = in + (size_t)(b * CIN + c) * Hin * Hin;
#pragma unroll
        for (int kh = 0; kh < 5; ++kh) {
            const float* row = inp + (size_t)iys[kh] * Hin;
#pragma unroll
            for (int kw = 0; kw < 5; ++kw) {
                float v = row[ixs[kw]];
#pragma unroll
                for (int o = 0; o < COUT; ++o)
                    acc[o] += v * ws[(o * CIN + c) * 25 + kh * 5 + kw];
            }
        }
    }

    size_t obase = (size_t)b * COUT * Hout * Hout + (size_t)oy * Hout + ox;
#pragma unroll
    for (int o = 0; o < COUT; ++o) {
        float r = acc[o] * As[o] + Bs[o];
        out[obase + (size_t)o * Hout * Hout] = fmaxf(r, 0.f);
    }
}

// ---------------------------------------------------------------------------
// Fused PatchMerging + q/k projection + windowed q.k^T attention logits.
// One workgroup (4 waves) per (batch, window). All GEMMs via
// V_WMMA_F32_16X16X4_F32 (exact fp32). LDS staging with padded strides
// (stride % 32 == 4 dwords) so all A/B ds_load_b64 fragments and the
// epilogue stores are LDS-bank-conflict free:
//   xm[64][36] -> x1[64][68] -> qs[64][132], ks[64][132] -> dots tiles
// ---------------------------------------------------------------------------
#define ATT_SCALE 0.17677669529663689f  /* 32^-0.5 */
#define XM_S 36
#define X1_S 68
#define QK_S 132

__global__ void __launch_bounds__(128) attn_kernel(
    const float* __restrict__ x3,     // [16][8][128][128]
    const float* __restrict__ pm_w,   // [64][32]
    const float* __restrict__ pm_b,   // [64]
    const float* __restrict__ qkv_w,  // [384][64] (only rows 0..255 used)
    const float* __restrict__ pos,    // [15][15]
    float* __restrict__ dots,         // [16][2][64][64][64]
    float* __restrict__ accg)         // [16][2] sigmoid sums
{
    extern __shared__ float sm[];
    float* xm   = sm;              // 64*36 = 2304: [token][feat32]
    float* x1   = sm + 2304;       // 64*68 = 4352: [token][64]
    float* qs   = sm + 6656;       // 64*132 = 8448: [token][128] (h*64+d)
    float* ks   = sm + 15104;      // 64*132 = 8448: [token][128] (h*64+d)
    float* ps   = sm + 23552;      // 225: rel-pos table
    float* sacc = sm + 23778;      // 2: per-head sigmoid partials

    const int blk = blockIdx.x;
    const int b   = blk >> 6;
    const int win = blk & 63;
    const int wy  = win >> 3, wx = win & 7;

    const int lane = threadIdx.x & 31;
    const int wv   = threadIdx.x >> 5;
    const int m    = lane & 15;         // row slot (A) == col slot (B/C)
    const int khi  = lane >> 4;         // K-half / M-half selector
    const int kh2  = khi << 1;          // K offset 0 or 2

    for (int i = threadIdx.x; i < 225; i += 128) ps[i] = pos[i];
    if (threadIdx.x < 2) sacc[threadIdx.x] = 0.f;

    // ---- stage xm: unfold(2,2) of conv3 output, feature order (c, kh, kw)
    for (int e = threadIdx.x; e < 2048; e += 128) {
        int t = e >> 5, f = e & 31;
        int c = f >> 2, ky = (f >> 1) & 1, kx = f & 1;
        int gy = wy * 8 + (t >> 3);
        int gx = wx * 8 + (t & 7);
        xm[t * XM_S + f] =
            x3[(((size_t)b * 8 + c) * 128 + 2 * gy + ky) * 128 + 2 * gx + kx];
    }
    __syncthreads();

    // ---- GEMM1: x1[64][64] = xm[64][32] @ pm_w^T + pm_b
    const int mt = wv;  // each wave owns one 16-row M tile
    for (int nt = 0; nt < 4; ++nt) {
        v8f acc = {};
        for (int k0 = 0; k0 < 32; k0 += 4) {
            int ka = k0 + kh2;
            v2f a = *(const v2f*)&xm[(mt * 16 + m) * XM_S + ka];
            v2f bf = *(const v2f*)(pm_w + (nt * 16 + m) * 32 + ka);
            acc = WMMA_F32_K4(a, bf, acc);
        }
        float bb = pm_b[nt * 16 + m];
#pragma unroll
        for (int r = 0; r < 8; ++r)
            x1[(mt * 16 + r + khi * 8) * X1_S + nt * 16 + m] = acc[r] + bb;
    }
    __syncthreads();

    // ---- GEMM2: [q|k][64][256] = x1 @ qkv_w[0:256]^T (q and k, row layout)
    for (int nt = 0; nt < 16; ++nt) {
        v8f acc = {};
        for (int k0 = 0; k0 < 64; k0 += 4) {
            int ka = k0 + kh2;
            v2f a = *(const v2f*)&x1[(mt * 16 + m) * X1_S + ka];
            v2f bf = *(const v2f*)(qkv_w + (nt * 16 + m) * 64 + ka);
            acc = WMMA_F32_K4(a, bf, acc);
        }
        int n = nt * 16 + m;
#pragma unroll
        for (int r = 0; r < 8; ++r) {
            int row = mt * 16 + r + khi * 8;
            float v = acc[r];
            if (n < 128) qs[row * QK_S + n] = v;
            else         ks[row * QK_S + (n - 128)] = v;
        }
    }
    __syncthreads();

    // ---- GEMM3: per head, dots = q_h @ k_h^T * scale + relpos; sigmoid sums
    //      B[d][j] = ks[j][h*64+d]: (d, d+1) adjacent -> single ds_load_b64
    for (int tt = 0; tt < 8; ++tt) {
        int tile = wv * 8 + tt;           // 32 tiles: h(2) x mt(4) x nt(4)
        int h   = tile >> 4;
        int mt3 = (tile >> 2) & 3;
        int nt3 = tile & 3;
        v8f acc = {};
        for (int k0 = 0; k0 < 64; k0 += 4) {
            int ka = k0 + kh2;
            v2f a  = *(const v2f*)&qs[(mt3 * 16 + m) * QK_S + h * 64 + ka];
            v2f bf = *(const v2f*)&ks[(nt3 * 16 + m) * QK_S + h * 64 + ka];
            acc = WMMA_F32_K4(a, bf, acc);
        }
        float ssum = 0.f;
        float* dptr = dots + (((size_t)b * 2 + h) * 64 + win) * 4096;
#pragma unroll
        for (int r = 0; r < 8; ++r) {
            int i = mt3 * 16 + r + khi * 8;   // query token
            int j = nt3 * 16 + m;             // key token
            float v = acc[r] * ATT_SCALE +
                      ps[((j >> 3) - (i >> 3) + 7) * 15 + ((j & 7) - (i & 7) + 7)];
            dptr[i * 64 + j] = v;
            ssum += 1.0f / (1.0f + __expf(-v));
        }
        atomicAdd(&sacc[h], ssum);           // ds_add_f32
    }
    __syncthreads();
    if (threadIdx.x < 2)
        atomicAdd(&accg[b * 2 + threadIdx.x], sacc[threadIdx.x]);
}

__global__ void zero_acc_kernel(float* a) {
    if (threadIdx.x < 32) a[threadIdx.x] = 0.f;
}

__global__ void finalize_kernel(const float* __restrict__ accg,
                                float* __restrict__ out) {
    int b = threadIdx.x;
    if (b < 16) {
        const float inv = 1.0f / 262144.0f;  // 64 windows * 64 * 64
        out[b]      = 3.0f * accg[b * 2 + 0] * inv;  // beta  (head 0)
        out[16 + b] = accg[b * 2 + 1] * inv;         // gamma (head 1)
    }
}

// ---------------------------------------------------------------------------
// Launch
// ---------------------------------------------------------------------------
extern "C" void kernel_launch(void* const* d_in, const int* in_sizes, int n_in,
                              void* d_out, int out_size, void* d_ws, size_t ws_size,
                              hipStream_t stream) {
    (void)in_sizes; (void)n_in; (void)out_size; (void)ws_size;
    // setup_inputs() order: x, enc_w[3], enc_b[3], bn_g[3], bn_b[3], bn_m[3],
    //                       bn_v[3], pm_w, pm_b, qkv_w, pos
    const float* x     = (const float*)d_in[0];
    const float* ew[3] = {(const float*)d_in[1], (const float*)d_in[2], (const float*)d_in[3]};
    const float* eb[3] = {(const float*)d_in[4], (const float*)d_in[5], (const float*)d_in[6]};
    const float* bg[3] = {(const float*)d_in[7], (const float*)d_in[8], (const float*)d_in[9]};
    const float* bb[3] = {(const float*)d_in[10], (const float*)d_in[11], (const float*)d_in[12]};
    const float* bm[3] = {(const float*)d_in[13], (const float*)d_in[14], (const float*)d_in[15]};
    const float* bv[3] = {(const float*)d_in[16], (const float*)d_in[17], (const float*)d_in[18]};
    const float* pm_w  = (const float*)d_in[19];
    const float* pm_b  = (const float*)d_in[20];
    const float* qkv_w = (const float*)d_in[21];
    const float* pos   = (const float*)d_in[22];

    float* out  = (float*)d_out;
    float* ws   = (float*)d_ws;
    float* c1   = ws;                                // 16*4*512*512 = 16,777,216 f
    float* c2   = ws + 16777216;                     // 16*4*256*256 =  4,194,304 f
    float* accg = ws + 16777216 + 4194304;           // 32 f
    float* x3   = out + 32;                          // [16][8][128][128]
    float* dots = out + 32 + 2097152;                // [16][2][64][64][64]

    zero_acc_kernel<<<1, 32, 0, stream>>>(accg);

    enc_block_kernel<3, 4><<<(16 * 512 * 512) / 256, 256, 0, stream>>>(
        x, ew[0], eb[0], bg[0], bb[0], bm[0], bv[0], c1, 16, 1024, 512);
    enc_block_kernel<4, 4><<<(16 * 256 * 256) / 256, 256, 0, stream>>>(
        c1, ew[1], eb[1], bg[1], bb[1], bm[1], bv[1], c2, 16, 512, 256);
    enc_block_kernel<4, 8><<<(16 * 128 * 128) / 256, 256, 0, stream>>>(
        c2, ew[2], eb[2], bg[2], bb[2], bm[2], bv[2], x3, 16, 256, 128);

    // 16 batches x 64 windows, 4 waves each; ~93 KB dynamic LDS (320 KB/WGP)
    attn_kernel<<<16 * 64, 128, 23784 * sizeof(float), stream>>>(
        x3, pm_w, pm_b, qkv_w, pos, dots, accg);

    finalize_kernel<<<1, 16, 0, stream>>>(accg, out);
}